// LinearExponentialDelayed_78245714198801
// MI455X (gfx1250) — compile-verified
//
#include <hip/hip_runtime.h>

// LinearExponentialDelayed step — elementwise, HBM-bound (~671 MB/call -> ~29us @ 23.3 TB/s).
// No live matmul in the reference (delay_init @ x feeds only a deleted value), so no WMMA:
// the optimal gfx1250 path here is streaming B128 loads/stores with non-temporal hints.

typedef float v4f __attribute__((ext_vector_type(4)));
typedef int   v4i __attribute__((ext_vector_type(4)));

namespace {
constexpr unsigned kOut = 8192;
constexpr unsigned kIn  = 4096;
constexpr unsigned kIn4 = kIn / 4;                 // float4s per row
constexpr unsigned kN4  = kOut * kIn4;             // 8,388,608 float4s per matrix
}

__global__ __launch_bounds__(256)
void led_step_kernel(const int*   __restrict__ x,        // (IN,1) binary spikes
                     const float* __restrict__ weight,   // (OUT,IN)
                     const float* __restrict__ trace,    // (OUT,IN)
                     const float* __restrict__ delay,    // (OUT,IN)
                     const float* __restrict__ sc_dt,
                     const float* __restrict__ sc_tau,
                     const float* __restrict__ sc_alpha,
                     float*       __restrict__ out_wspike, // (OUT,IN) = weight * spike_out
                     float*       __restrict__ out_trace)  // (OUT,IN) = trace_new
{
    // Uniform scalar loads -> SMEM path.
    const float k     = sc_dt[0] / sc_tau[0];
    const float alpha = sc_alpha[0];

    const unsigned idx4 = blockIdx.x * blockDim.x + threadIdx.x;
    if (idx4 >= kN4) return;

    const unsigned col4 = idx4 & (kIn4 - 1u);      // column group within the row

    // x: 16 KB, reused by all 8192 rows -> regular-temporal (stays in WGP$/L2).
    const v4i xi = *reinterpret_cast<const v4i*>(x + 4u * col4);

    // Streamed operands: each byte read exactly once -> non-temporal B128 loads
    // (global_load_b128 th:NT) so the 671 MB stream does not evict useful L2 lines.
    const v4f wv = __builtin_nontemporal_load(reinterpret_cast<const v4f*>(weight) + idx4);
    const v4f tv = __builtin_nontemporal_load(reinterpret_cast<const v4f*>(trace)  + idx4);
    const v4f dv = __builtin_nontemporal_load(reinterpret_cast<const v4f*>(delay)  + idx4);

    v4f ow, ot;
#pragma unroll
    for (int j = 0; j < 4; ++j) {
        const float xf = (float)xi[j];
        // trace_new = trace + (dt/tau) * (-trace + alpha * x)
        ot[j] = tv[j] + k * (alpha * xf - tv[j]);
        // propagate_spike: decrement positive countdowns, spike where it hits 1
        const float d  = dv[j];
        const float d1 = (d > 0.0f) ? (d - 1.0f) : d;
        ow[j] = (d1 == 1.0f) ? wv[j] : 0.0f;       // weight * spike_out
    }

    // Outputs are write-once streams -> non-temporal B128 stores.
    __builtin_nontemporal_store(ow, reinterpret_cast<v4f*>(out_wspike) + idx4);
    __builtin_nontemporal_store(ot, reinterpret_cast<v4f*>(out_trace)  + idx4);
}

extern "C" void kernel_launch(void* const* d_in, const int* in_sizes, int n_in,
                              void* d_out, int out_size, void* d_ws, size_t ws_size,
                              hipStream_t stream) {
    const int*   x      = (const int*)  d_in[0];
    const float* weight = (const float*)d_in[1];
    const float* trace  = (const float*)d_in[2];
    const float* delay  = (const float*)d_in[3];
    // d_in[4] = delay_init: feeds only delay_new, which the reference deletes -> never read.
    const float* sdt    = (const float*)d_in[5];
    const float* stau   = (const float*)d_in[6];
    const float* salpha = (const float*)d_in[7];

    float* out_wspike = (float*)d_out;                              // first output, flat
    float* out_trace  = (float*)d_out + (size_t)kOut * (size_t)kIn; // second output, flat

    const unsigned threads = 256;                    // 8 wave32 waves per block
    const unsigned blocks  = (kN4 + threads - 1u) / threads; // 32768 blocks

    led_step_kernel<<<blocks, threads, 0, stream>>>(
        x, weight, trace, delay, sdt, stau, salpha, out_wspike, out_trace);
}